// KSubspaceBaseModel_67439576482149
// MI455X (gfx1250) — compile-verified
//
#include <hip/hip_runtime.h>
#include <hip/hip_bf16.h>

// Problem constants (from reference)
#define R_ 8
#define K_ 32
#define B_ 1024
#define D_ 256
#define L_ 16
#define EMA 0.9f

typedef __attribute__((ext_vector_type(2))) float v2f;
typedef __attribute__((ext_vector_type(8))) float v8f;

// -----------------------------------------------------------------------------
// Kernel 1: per-(r,k) GEMM  x_ = z @ Us^T  via V_WMMA_F32_16X16X4_F32,
// fused with loss[b,r,k] = 0.5 * sum_D (x - x_)^2.
// One wave handles one (r,k) pair and a 16-row b-tile -> a 16x256 output strip.
// 16384 waves total = 2048 blocks x 8 waves.
// -----------------------------------------------------------------------------
__global__ __launch_bounds__(256) void ks_gemm_loss_kernel(
    const float* __restrict__ x,    // [B][D]
    const float* __restrict__ z,    // [R*K][B][L]
    const float* __restrict__ Us,   // [R*K][D][L]
    float* __restrict__ xhat,       // [R*K][B][D]
    float* __restrict__ loss_ws)    // [B][R][K]
{
    const int lane = threadIdx.x & 31;
    const int wid  = blockIdx.x * 8 + (threadIdx.x >> 5);   // 0..16383
    const int rk   = wid >> 6;                              // 0..255  (r*K + k)
    const int b0   = (wid & 63) << 4;                       // b tile base
    const int half = lane >> 4;                             // 0: lanes 0-15, 1: lanes 16-31
    const int l16  = lane & 15;

    const float* zrk   = z  + ((size_t)rk * B_ + b0) * L_;
    const float* Urk   = Us + (size_t)rk * D_ * L_;
    float*       outrk = xhat + ((size_t)rk * B_ + b0) * D_;

    // --- Load A (z tile, 16x16) into 16x4-f32 WMMA layout, 4 K-steps of 4.
    // Lane L<16 : row=b0+L,  holds (K=4kk+0, 4kk+1) ; lanes>=16: (K=4kk+2, 4kk+3)
    v2f a[4];
    {
        const float* zrow = zrk + (size_t)l16 * L_ + 2 * half;
        #pragma unroll
        for (int kk = 0; kk < 4; ++kk)
            a[kk] = *(const v2f*)(zrow + 4 * kk);
    }

    float lsum[8];
    #pragma unroll
    for (int i = 0; i < 8; ++i) lsum[i] = 0.0f;

    // --- Sweep the 16 column tiles (D = 256 = 16 * 16)
    for (int nt = 0; nt < 16; ++nt) {
        const int n0 = nt << 4;

        // B matrix (4x16 = Us^T slice): lane holds column n=l16,
        // rows (4kk+0,4kk+1) in lanes 0-15, (4kk+2,4kk+3) in lanes 16-31.
        // B[k_row][n] = Us[n0+n][k_row]  -> contiguous float2 in Us row.
        v2f bm[4];
        const float* urow = Urk + (size_t)(n0 + l16) * L_ + 2 * half;
        #pragma unroll
        for (int kk = 0; kk < 4; ++kk)
            bm[kk] = *(const v2f*)(urow + 4 * kk);

        v8f acc = {};
        #pragma unroll
        for (int kk = 0; kk < 4; ++kk)
            acc = __builtin_amdgcn_wmma_f32_16x16x4_f32(
                /*neg_a=*/false, a[kk], /*neg_b=*/false, bm[kk],
                /*c_mod=*/(short)0, acc, /*reuse_a=*/false, /*reuse_b=*/false);

        // Epilogue: C/D layout -> VGPR i, lanes 0-15: row i, col lane;
        //                         lanes 16-31: row 8+i, col lane-16.
        const int col     = n0 + l16;
        const int rowbase = half << 3;
        #pragma unroll
        for (int i = 0; i < 8; ++i) {
            const float v = acc[i];
            // 268 MB output stream > 192 MB L2 -> non-temporal store
            __builtin_nontemporal_store(v, outrk + (size_t)(rowbase + i) * D_ + col);
            const float d = v - x[(size_t)(b0 + rowbase + i) * D_ + col];
            lsum[i] += d * d;
        }
    }

    // --- Reduce the per-lane column partials over the 16 lanes of each half.
    // xor masks 1,2,4,8 never cross the 16-lane boundary (wave32).
    #pragma unroll
    for (int i = 0; i < 8; ++i) {
        float s = lsum[i];
        s += __shfl_xor(s, 1, 32);
        s += __shfl_xor(s, 2, 32);
        s += __shfl_xor(s, 4, 32);
        s += __shfl_xor(s, 8, 32);
        lsum[i] = s;
    }
    if (l16 == 0) {  // lane 0 -> rows b0..b0+7, lane 16 -> rows b0+8..b0+15
        const int r = rk >> 5;
        const int k = rk & 31;
        #pragma unroll
        for (int i = 0; i < 8; ++i) {
            const int b = b0 + (half << 3) + i;
            loss_ws[((size_t)b * R_ + r) * K_ + k] = 0.5f * lsum[i];
        }
    }
}

// -----------------------------------------------------------------------------
// Kernel 2: per-(b,r) top-2-over-K assignment. One wave per (b,r), lane = k.
// -----------------------------------------------------------------------------
__global__ __launch_bounds__(256) void ks_assign_kernel(
    const float* __restrict__ loss_ws,  // [B][R][K]
    float* __restrict__ c_out,          // [B][R][K]
    float* __restrict__ cnt_ws,         // [R][K]
    float* __restrict__ val_ws,         // [R][K]
    float* __restrict__ obj_ws)         // [R]
{
    const int lane = threadIdx.x & 31;                     // k
    const int wid  = blockIdx.x * 8 + (threadIdx.x >> 5);  // 0..8191
    const int b    = wid >> 3;
    const int r    = wid & 7;

    const float val = loss_ws[((size_t)b * R_ + r) * K_ + lane];

    // wave min
    float m1 = val;
    #pragma unroll
    for (int m = 16; m >= 1; m >>= 1) m1 = fminf(m1, __shfl_xor(m1, m, 32));

    // argmin: lowest lane index among minima (matches top_k tie-breaking)
    const unsigned long long ball = __ballot(val == m1);
    const int idx1 = __ffsll(ball) - 1;

    // second minimum (exclude idx1)
    float m2 = (lane == idx1) ? 3.402823466e+38f : val;
    #pragma unroll
    for (int m = 16; m >= 1; m >>= 1) m2 = fminf(m2, __shfl_xor(m2, m, 32));

    // one-hot c (coalesced: lane k writes c[b,r,k])
    c_out[((size_t)b * R_ + r) * K_ + lane] = (lane == idx1) ? 1.0f : 0.0f;

    if (lane == 0) {
        atomicAdd(&cnt_ws[r * K_ + idx1], 1.0f);
        atomicAdd(&val_ws[r * K_ + idx1], m2 - m1);
        atomicAdd(&obj_ws[r], m1);
    }
}

// -----------------------------------------------------------------------------
// Kernel 0: zero the accumulator region of the workspace (harness poisons ws).
// -----------------------------------------------------------------------------
__global__ void ks_zero_kernel(float* __restrict__ p, int n)
{
    const int i = blockIdx.x * blockDim.x + threadIdx.x;
    if (i < n) p[i] = 0.0f;
}

// -----------------------------------------------------------------------------
// Kernel 3: finalize EMA stats, obj, obj_mean.
// -----------------------------------------------------------------------------
__global__ __launch_bounds__(256) void ks_finalize_kernel(
    const float* __restrict__ c_mean_in,  // [R*K]
    const float* __restrict__ value_in,   // [R*K]
    const float* __restrict__ cnt_ws,     // [R*K]
    const float* __restrict__ val_ws,     // [R*K]
    const float* __restrict__ obj_ws,     // [R]
    float* __restrict__ obj_out,          // [R]
    float* __restrict__ obj_mean_out,     // [1]
    float* __restrict__ ncm_out,          // [R*K]
    float* __restrict__ nv_out)           // [R*K]
{
    const int t = threadIdx.x;            // 0..255 = r*K + k
    const float invB = 1.0f / (float)B_;
    ncm_out[t] = EMA * c_mean_in[t] + (1.0f - EMA) * (cnt_ws[t] * invB);
    nv_out[t]  = EMA * value_in[t]  + (1.0f - EMA) * (val_ws[t] * invB);
    if (t < R_) obj_out[t] = obj_ws[t] * invB;
    if (t == 0) {
        float s = 0.0f;
        for (int r = 0; r < R_; ++r) s += obj_ws[r];
        obj_mean_out[0] = s * invB / (float)R_;
    }
}

// -----------------------------------------------------------------------------
// Host launcher
// -----------------------------------------------------------------------------
extern "C" void kernel_launch(void* const* d_in, const int* in_sizes, int n_in,
                              void* d_out, int out_size, void* d_ws, size_t ws_size,
                              hipStream_t stream)
{
    (void)in_sizes; (void)n_in; (void)out_size; (void)ws_size;

    const float* x      = (const float*)d_in[0];  // [B][D]
    const float* z      = (const float*)d_in[1];  // [R][K][B][L]
    const float* Us     = (const float*)d_in[2];  // [R][K][D][L]
    const float* c_mean = (const float*)d_in[3];  // [R][K]
    const float* value  = (const float*)d_in[4];  // [R][K]

    float* out = (float*)d_out;
    // Output layout (flat, in tuple-return order):
    float* xhat_out     = out;                                    // 8*32*1024*256
    float* c_out        = out + (size_t)R_ * K_ * B_ * D_;        // 1024*8*32
    float* obj_out      = c_out + (size_t)B_ * R_ * K_;           // 8
    float* obj_mean_out = obj_out + R_;                           // 1
    float* ncm_out      = obj_mean_out + 1;                       // 256
    float* nv_out       = ncm_out + (size_t)R_ * K_;              // 256

    // Workspace layout
    float* loss_ws = (float*)d_ws;                        // B*R*K = 262144 floats
    float* cnt_ws  = loss_ws + (size_t)B_ * R_ * K_;      // 256
    float* val_ws  = cnt_ws + R_ * K_;                    // 256
    float* obj_ws  = val_ws + R_ * K_;                    // 8

    // 0) zero accumulators (R*K*2 + R = 520 floats)
    ks_zero_kernel<<<3, 256, 0, stream>>>(cnt_ws, 2 * R_ * K_ + R_);

    // 1) batched WMMA GEMM + fused loss: 16384 waves = 2048 blocks x 8 waves
    ks_gemm_loss_kernel<<<2048, 256, 0, stream>>>(x, z, Us, xhat_out, loss_ws);

    // 2) top-2 assignment: 8192 waves = 1024 blocks x 8 waves
    ks_assign_kernel<<<1024, 256, 0, stream>>>(loss_ws, c_out, cnt_ws, val_ws, obj_ws);

    // 3) finalize EMA / obj / obj_mean
    ks_finalize_kernel<<<1, 256, 0, stream>>>(c_mean, value, cnt_ws, val_ws, obj_ws,
                                              obj_out, obj_mean_out, ncm_out, nv_out);
}